// ShenzhenGeometry_74062416053309
// MI455X (gfx1250) — compile-verified
//
#include <hip/hip_runtime.h>
#include <stdint.h>

// ---------------------------------------------------------------------------
// Geometry constants (match reference)
// ---------------------------------------------------------------------------
#define Zs    160
#define Ys    224
#define Xs    224
#define NAng  90
#define DETC  224
#define DETR  160          // == Zs
#define NT    224
#define ACHUNK 15          // angles per forward workgroup
#define NCHUNK 6           // 6 * 15 == 90
#define CTRF  111.5f       // (224-1)/2 : cx, cy, det-center, step-center
#define DTH   0.0349065850398865915f  // pi / 90

typedef float v2f  __attribute__((ext_vector_type(2)));
typedef float v8f  __attribute__((ext_vector_type(8)));
typedef unsigned int u32x4 __attribute__((ext_vector_type(4)));
typedef int   i32x4 __attribute__((ext_vector_type(4)));
typedef int   i32x8 __attribute__((ext_vector_type(8)));

// --- feature probes (device pass only; host pass falls back so AST stays valid)
#if defined(__HIP_DEVICE_COMPILE__) && defined(__has_builtin)
#if __has_builtin(__builtin_amdgcn_wmma_f32_16x16x4_f32)
#define HAVE_WMMA4 1
#endif
#if __has_builtin(__builtin_amdgcn_tensor_load_to_lds) && __has_builtin(__builtin_amdgcn_s_wait_tensorcnt)
#define HAVE_TDM 1
#endif
#endif
#ifndef HAVE_WMMA4
#define HAVE_WMMA4 0
#endif
#ifndef HAVE_TDM
#define HAVE_TDM 0
#endif

// ---------------------------------------------------------------------------
// Branchless bilinear sample from the LDS-resident z-slice. Indices are
// clamped so the DS loads are always in-bounds; the reference's "full 2x2
// footprint in-bounds" rule is applied as a final select. No exec-mask
// divergence, so the surrounding WMMA always sees EXEC all-ones.
// ---------------------------------------------------------------------------
__device__ __forceinline__ float bilin(const float* sl, float xs, float ys) {
    float x0 = floorf(xs), y0 = floorf(ys);
    float fx = xs - x0,    fy = ys - y0;
    int xi = (int)x0, yi = (int)y0;
    const bool inb = ((unsigned)xi <= (unsigned)(Xs - 2)) &
                     ((unsigned)yi <= (unsigned)(Ys - 2));
    int xc = xi < 0 ? 0 : (xi > Xs - 2 ? Xs - 2 : xi);
    int yc = yi < 0 ? 0 : (yi > Ys - 2 ? Ys - 2 : yi);
    const float* r0 = sl + yc * Xs + xc;
    float v00 = r0[0],  v01 = r0[1];
    float v10 = r0[Xs], v11 = r0[Xs + 1];
    float a0 = v00 + fx * (v01 - v00);
    float a1 = v10 + fx * (v11 - v10);
    float v  = a0 + fy * (a1 - a0);
    return inb ? v : 0.0f;
}

// ---------------------------------------------------------------------------
// Kernel 1: forward projection + residual.
// Grid (Zs, NCHUNK), block 448 (14 waves). Each WG owns one z-slice (196 KB,
// staged once into LDS by the Tensor Data Mover) and ACHUNK angles. Wave w
// owns u in [16w, 16w+16); the sum over t runs on V_WMMA_F32_16X16X4_F32
// with an all-ones B (exact row-sum accumulate into C).
// ---------------------------------------------------------------------------
__global__ __launch_bounds__(448) void fwd_res_kernel(
        const float* __restrict__ vol, const float* __restrict__ p,
        float* __restrict__ res) {
    __shared__ float slice[Ys * Xs];   // 196 KB of the 320 KB WGP LDS
    const int z     = blockIdx.x;
    const int chunk = blockIdx.y;
    const float* gsrc = vol + (size_t)z * (Ys * Xs);

#if HAVE_TDM
    // Opaque zero: the compiler cannot prove the guarded store below never
    // runs, so reads of 'slice' (really written by the TDM behind the
    // descriptor) cannot be folded to undef. At runtime junk == 0.
    unsigned junk;
    asm volatile("s_mov_b32 %0, 0" : "=s"(junk));
    if (junk != 0u) {
        for (int i = threadIdx.x; i < Ys * Xs; i += 448)
            slice[i] = (float)junk;
    }
    if (threadIdx.x == 0) {
        unsigned long long ga = (unsigned long long)(uintptr_t)gsrc;
        // Low 32 bits of the flat shared-aperture address == LDS byte offset.
        const unsigned lds_off = (unsigned)(uintptr_t)(void*)&slice[0];
        // D# group 0: count=1 (valid), lds_addr, 57-bit global addr, type=2.
        u32x4 g0;
        g0.x = 1u;
        g0.y = lds_off;
        g0.z = (unsigned)(ga & 0xffffffffull);
        g0.w = ((unsigned)((ga >> 32) & 0x01ffffffull)) | 0x80000000u;
        // D# group 1: data_size=4B; 1-D tile: tensor_dim0 = tile_dim0 = 50176,
        // tensor_dim1 = tile_dim1 = 1, stride0 = 50176.
        i32x8 g1;
        g1[0] = 0x00020000;        // wg_mask=0, data_size=2 (4 bytes)
        g1[1] = (int)0xC4000000;   // tensor_dim0[15:0]=50176 in bits[31:16]
        g1[2] = 0x00010000;        // tensor_dim0 hi=0 ; tensor_dim1=1
        g1[3] = (int)0xC4000000;   // tile_dim0 = 50176
        g1[4] = 1;                 // tile_dim1 = 1, tile_dim2 = 0
        g1[5] = 50176;             // tensor_dim0_stride lo
        g1[6] = 0;
        g1[7] = 0;
        i32x4 gz = {0, 0, 0, 0};
#if __clang_major__ >= 23
        i32x8 g4 = {0, 0, 0, 0, 0, 0, 0, 0};
        __builtin_amdgcn_tensor_load_to_lds(g0, g1, gz, gz, g4, 0);
#else
        __builtin_amdgcn_tensor_load_to_lds(g0, g1, gz, gz, 0);
#endif
    }
    __builtin_amdgcn_s_wait_tensorcnt(0);
#else
    for (int i = threadIdx.x; i < Ys * Xs; i += 448) slice[i] = gsrc[i];
#endif
    __syncthreads();

    const int lane = threadIdx.x & 31;
    const int wave = threadIdx.x >> 5;        // 0..13 -> u group
    const int m    = lane & 15;               // A-matrix row (u offset)
    const int half = lane >> 4;               // lanes 16-31 hold K=2,3
    const int u0   = wave * 16;
    const float uf = (float)(u0 + m) - CTRF;

    for (int ai = 0; ai < ACHUNK; ++ai) {
        const int a = chunk * ACHUNK + ai;
        const float th = (float)a * DTH;
        const float s = sinf(th), c = cosf(th);
        // lane's first t index within each 4-step block is tb + 2*half
        float xs = CTRF - uf * s + ((float)(2 * half) - CTRF) * c;
        float ys = CTRF + uf * c + ((float)(2 * half) - CTRF) * s;
#if HAVE_WMMA4
        v8f acc = {0.f, 0.f, 0.f, 0.f, 0.f, 0.f, 0.f, 0.f};
        const v2f bones = {1.0f, 1.0f};
#else
        float psum = 0.0f;
#endif
        for (int tb = 0; tb < NT; tb += 4) {
            float s0 = bilin(slice, xs, ys);
            float s1 = bilin(slice, xs + c, ys + s);
#if HAVE_WMMA4
            v2f am; am.x = s0; am.y = s1;   // A 16x4 f32: K = 2*half + {0,1}
            acc = __builtin_amdgcn_wmma_f32_16x16x4_f32(
                false, am, false, bones, (short)0, acc, false, false);
#else
            psum += s0 + s1;
#endif
            xs += 4.0f * c;
            ys += 4.0f * s;
        }
        const int base = (a * DETR + z) * DETC + u0;
#if HAVE_WMMA4
        // C/D layout: lane 0 holds rows 0..7 (one per VGPR), lane 16 rows 8..15
        if (lane == 0) {
            for (int r = 0; r < 8; ++r) res[base + r] = acc[r] - p[base + r];
        } else if (lane == 16) {
            for (int r = 0; r < 8; ++r) res[base + 8 + r] = acc[r] - p[base + 8 + r];
        }
#else
        float tot = psum + __shfl_xor(psum, 16, 32);
        if (lane < 16) res[base + m] = tot - p[base + m];
#endif
    }
}

// ---------------------------------------------------------------------------
// Kernel 2: exact adjoint (gather form, atomic-free, deterministic) fused
// with out = x - lamb*bp. The sample lattice per angle is a rotated unit
// lattice, so voxel (x,y) receives from the <=3x3 (u,t) lattice points whose
// bilinear footprint covers it; candidates found by inverse rotation.
// One block = 256 consecutive voxels of a single z-slice (196 blocks / z).
// ---------------------------------------------------------------------------
__global__ __launch_bounds__(256) void bp_axpy_kernel(
        const float* __restrict__ xin, const float* __restrict__ res,
        const float* __restrict__ lambp, float* __restrict__ out) {
    __shared__ float rrow[DETC];
    __shared__ float ssin[NAng];
    __shared__ float scos[NAng];
    const int tid = threadIdx.x;
    const int z   = blockIdx.x / 196;
    const int rem = (blockIdx.x % 196) * 256 + tid;
    const int y   = rem / Xs;
    const int xq  = rem - y * Xs;
    if (tid < NAng) {
        float th = (float)tid * DTH;
        ssin[tid] = sinf(th);
        scos[tid] = cosf(th);
    }
    __syncthreads();
    const float dx = (float)xq - CTRF;
    const float dy = (float)y  - CTRF;
    float acc = 0.0f;
    for (int a = 0; a < NAng; ++a) {
        if (tid < DETC) {
            rrow[tid] = res[(a * DETR + z) * DETC + tid];
            if (a + 1 < NAng)  // pull next row toward L0 while we compute
                __builtin_prefetch(&res[((a + 1) * DETR + z) * DETC + tid], 0, 0);
        }
        __syncthreads();
        const float s = ssin[a], c = scos[a];
        // inverse rotation (matrix [[-s,c],[c,s]] is its own inverse)
        const float iu0 = rintf(-dx * s + dy * c + CTRF);
        const float it0 = rintf( dx * c + dy * s + CTRF);
        for (int du = -1; du <= 1; ++du) {
            const float iuf = iu0 + (float)du;
            const int   iu  = (int)iuf;
            const float r   = (iu >= 0 && iu < DETC) ? rrow[iu] : 0.0f;
            const float usv = iuf - CTRF;
            const float bx  = CTRF - usv * s;
            const float by  = CTRF + usv * c;
            float wsum = 0.0f;
            for (int dt = -1; dt <= 1; ++dt) {
                const float itf = it0 + (float)dt;
                const float tsv = itf - CTRF;
                const float xsv = bx + tsv * c;
                const float ysv = by + tsv * s;
                const float ax  = fabsf(xsv - (float)xq);
                const float ay  = fabsf(ysv - (float)y);
                bool ok = (itf >= 0.0f) & (itf <= 223.0f) &
                          (xsv >= 0.0f) & (xsv < 223.0f) &
                          (ysv >= 0.0f) & (ysv < 223.0f) &
                          (ax < 1.0f) & (ay < 1.0f);
                wsum += ok ? (1.0f - ax) * (1.0f - ay) : 0.0f;
            }
            acc += wsum * r;
        }
        __syncthreads();
    }
    const size_t gi = (size_t)blockIdx.x * 256 + tid;
    out[gi] = xin[gi] - lambp[0] * acc;
}

// ---------------------------------------------------------------------------
extern "C" void kernel_launch(void* const* d_in, const int* in_sizes, int n_in,
                              void* d_out, int out_size, void* d_ws, size_t ws_size,
                              hipStream_t stream) {
    const float* x    = (const float*)d_in[0];  // [1,1,160,224,224]
    const float* p    = (const float*)d_in[1];  // [1,1,90,160,224]
    const float* lamb = (const float*)d_in[2];  // scalar
    float* out = (float*)d_out;                 // [1,1,160,224,224]
    float* res = (float*)d_ws;                  // residual sinogram: 12.9 MB

    fwd_res_kernel<<<dim3(Zs, NCHUNK), dim3(448), 0, stream>>>(x, p, res);
    bp_axpy_kernel<<<dim3(Zs * 196), dim3(256), 0, stream>>>(x, res, lamb, out);

    (void)in_sizes; (void)n_in; (void)out_size; (void)ws_size;
}